// SimilarClusterEncoder_73882027425984
// MI455X (gfx1250) — compile-verified
//
#include <hip/hip_runtime.h>

#define BT     16384
#define DIM    512
#define NK     4096
#define DOUT   1024
#define EPSF   1e-6f
#define LSTR   516            // LDS row stride in floats (bank-conflict-free)
#define MT     4              // M-tiles per block
#define ROWS   (MT * 16)      // 64 rows per block

typedef __attribute__((ext_vector_type(2))) float v2f;
typedef __attribute__((ext_vector_type(8))) float v8f;

typedef __attribute__((address_space(3))) char lds_char;

// ---- CDNA5 async global->LDS copy (ASYNCcnt path) ----
__device__ __forceinline__ unsigned lds_addr_of(const void* p) {
  return (unsigned)(unsigned long long)(lds_char*)p;
}
__device__ __forceinline__ void async_g2l_b128(unsigned lds_off, const void* gptr) {
  asm volatile("global_load_async_to_lds_b128 %0, %1, off"
               :: "v"(lds_off), "v"(gptr) : "memory");
}
__device__ __forceinline__ void wait_async0() {
  asm volatile("s_wait_asynccnt 0" ::: "memory");
}

// ---------------- Kernel 1: adjusted center norms ----------------
// c2p[k] = |c_k|^2 - 2*EPS*sum(c_k)   (folds the x+EPS shift out of the GEMM)
__global__ __launch_bounds__(256) void c2_kernel(const float* __restrict__ centers,
                                                 float* __restrict__ c2p) {
  int k = blockIdx.x * 256 + threadIdx.x;
  const float* row = centers + (size_t)k * DIM;
  float s = 0.f, t = 0.f;
  for (int d = 0; d < DIM; d += 4) {
    float4 v = *(const float4*)(row + d);
    s += v.x * v.x + v.y * v.y + v.z * v.z + v.w * v.w;
    t += v.x + v.y + v.z + v.w;
  }
  c2p[k] = s - 2.0f * EPSF * t;
}

// ---------------- Kernel 2: WMMA distance + argmin ----------------
// block = 256 threads (8 waves), owns 64 rows; each wave sweeps 32 N-tiles,
// each B fragment feeds 4 WMMAs (4 M-tiles).
__global__ __launch_bounds__(256) void argmin_kernel(const float* __restrict__ x,
                                                     const float* __restrict__ centers,
                                                     const float* __restrict__ c2p,
                                                     int* __restrict__ sel) {
  __shared__ float lsX[ROWS * LSTR];
  __shared__ float redV[8][ROWS];
  __shared__ int   redI[8][ROWS];

  const int tid  = threadIdx.x;
  const int wave = tid >> 5;
  const int lane = tid & 31;
  const int l15  = lane & 15;
  const int kh   = lane >> 4;
  const int rowBase = blockIdx.x * ROWS;

  // async-stage raw x rows into LDS (16B chunks stay inside a padded row)
  for (int i = tid; i < ROWS * (DIM / 4); i += 256) {
    int r  = i >> 7;            // 128 chunks per row
    int d4 = (i & 127) << 2;
    async_g2l_b128(lds_addr_of(&lsX[r * LSTR + d4]),
                   &x[(size_t)(rowBase + r) * DIM + d4]);
  }
  wait_async0();
  __syncthreads();

  float bestV[MT][8];
  int   bestI[MT][8];
#pragma unroll
  for (int m = 0; m < MT; ++m)
#pragma unroll
    for (int r = 0; r < 8; ++r) { bestV[m][r] = 3.4e38f; bestI[m][r] = 0; }

  const float* aB[MT];
#pragma unroll
  for (int m = 0; m < MT; ++m)
    aB[m] = &lsX[(m * 16 + l15) * LSTR + kh * 2];

  for (int t = 0; t < 32; ++t) {                  // 32 N-tiles per wave
    const int col = (wave * 32 + t) * 16 + l15;
    const float* bBase = &centers[(size_t)col * DIM + kh * 2];
    v8f acc[MT];
#pragma unroll
    for (int m = 0; m < MT; ++m) acc[m] = (v8f){0.f,0.f,0.f,0.f,0.f,0.f,0.f,0.f};

#pragma unroll 2
    for (int kk = 0; kk < DIM; kk += 4) {
      v2f b = *(const v2f*)(bBase + kk);
#pragma unroll
      for (int m = 0; m < MT; ++m) {
        v2f a = *(const v2f*)(aB[m] + kk);
        acc[m] = __builtin_amdgcn_wmma_f32_16x16x4_f32(false, a, false, b,
                                                       (short)0, acc[m], false, false);
      }
    }
    float cc = c2p[col];
#pragma unroll
    for (int m = 0; m < MT; ++m)
#pragma unroll
      for (int r = 0; r < 8; ++r) {
        float v = fmaf(-2.0f, acc[m][r], cc);     // score = c2' - 2*x.c
        if (v < bestV[m][r]) { bestV[m][r] = v; bestI[m][r] = col; }
      }
  }

  // reduce across the 16 lanes of each half-wave (C layout: lane = N)
#pragma unroll
  for (int off = 8; off >= 1; off >>= 1) {
#pragma unroll
    for (int m = 0; m < MT; ++m)
#pragma unroll
      for (int r = 0; r < 8; ++r) {
        float ov = __shfl_xor(bestV[m][r], off, 32);
        int   oi = __shfl_xor(bestI[m][r], off, 32);
        if (ov < bestV[m][r] || (ov == bestV[m][r] && oi < bestI[m][r])) {
          bestV[m][r] = ov; bestI[m][r] = oi;
        }
      }
  }
  if (l15 == 0) {
#pragma unroll
    for (int m = 0; m < MT; ++m)
#pragma unroll
      for (int r = 0; r < 8; ++r) {
        redV[wave][m * 16 + kh * 8 + r] = bestV[m][r];
        redI[wave][m * 16 + kh * 8 + r] = bestI[m][r];
      }
  }
  __syncthreads();
  if (tid < ROWS) {
    float bv = redV[0][tid]; int bi = redI[0][tid];
#pragma unroll
    for (int w = 1; w < 8; ++w) {
      float v = redV[w][tid]; int i2 = redI[w][tid];
      if (v < bv || (v == bv && i2 < bi)) { bv = v; bi = i2; }
    }
    sel[rowBase + tid] = bi;
  }
}

// ---------------- Kernel 3: gather + projection GEMM ----------------
// block owns 64 output rows; wave covers 128 cols in 4 groups of 2 N-tiles,
// 4 M-tiles x 2 N-tiles = 8 WMMAs per B-pair.
__global__ __launch_bounds__(256) void proj_kernel(const float* __restrict__ centers,
                                                   const float* __restrict__ W,
                                                   const float* __restrict__ bias,
                                                   const int* __restrict__ sel,
                                                   float* __restrict__ out) {
  __shared__ float lsQ[ROWS * LSTR];
  const int tid  = threadIdx.x;
  const int wave = tid >> 5;
  const int lane = tid & 31;
  const int l15  = lane & 15;
  const int kh   = lane >> 4;
  const int rowBase = blockIdx.x * ROWS;

  // async gather of selected center rows into LDS
  for (int i = tid; i < ROWS * (DIM / 4); i += 256) {
    int r  = i >> 7;
    int d4 = (i & 127) << 2;
    int srow = sel[rowBase + r];
    async_g2l_b128(lds_addr_of(&lsQ[r * LSTR + d4]),
                   &centers[(size_t)srow * DIM + d4]);
  }
  wait_async0();
  __syncthreads();

  const float* aB[MT];
#pragma unroll
  for (int m = 0; m < MT; ++m)
    aB[m] = &lsQ[(m * 16 + l15) * LSTR + kh * 2];

  for (int g = 0; g < 4; ++g) {
    const int colB = wave * 128 + g * 32;
    v8f acc[MT][2];
#pragma unroll
    for (int m = 0; m < MT; ++m) {
      acc[m][0] = (v8f){0.f,0.f,0.f,0.f,0.f,0.f,0.f,0.f};
      acc[m][1] = (v8f){0.f,0.f,0.f,0.f,0.f,0.f,0.f,0.f};
    }
    for (int kk = 0; kk < DIM; kk += 4) {
      const int krow = kk + kh * 2;
      v2f b0, b1;
      b0.x = W[(size_t)krow * DOUT + colB + l15];
      b0.y = W[(size_t)(krow + 1) * DOUT + colB + l15];
      b1.x = W[(size_t)krow * DOUT + colB + 16 + l15];
      b1.y = W[(size_t)(krow + 1) * DOUT + colB + 16 + l15];
#pragma unroll
      for (int m = 0; m < MT; ++m) {
        v2f a = *(const v2f*)(aB[m] + kk);
        acc[m][0] = __builtin_amdgcn_wmma_f32_16x16x4_f32(false, a, false, b0,
                                                          (short)0, acc[m][0], false, false);
        acc[m][1] = __builtin_amdgcn_wmma_f32_16x16x4_f32(false, a, false, b1,
                                                          (short)0, acc[m][1], false, false);
      }
    }
#pragma unroll
    for (int j = 0; j < 2; ++j) {
      int col = colB + j * 16 + l15;
      float bb = bias[col];
#pragma unroll
      for (int m = 0; m < MT; ++m)
#pragma unroll
        for (int r = 0; r < 8; ++r) {
          int row = rowBase + m * 16 + r + kh * 8;
          out[(size_t)row * DOUT + col] = acc[m][j][r] + bb;
        }
    }
  }
}

extern "C" void kernel_launch(void* const* d_in, const int* in_sizes, int n_in,
                              void* d_out, int out_size, void* d_ws, size_t ws_size,
                              hipStream_t stream) {
  const float* x       = (const float*)d_in[0];
  const float* centers = (const float*)d_in[1];
  const float* W       = (const float*)d_in[2];
  const float* b       = (const float*)d_in[3];
  float* out = (float*)d_out;

  float* c2p = (float*)d_ws;                              // 4096 f32
  int*   sel = (int*)((char*)d_ws + NK * sizeof(float));  // 16384 i32

  c2_kernel<<<NK / 256, 256, 0, stream>>>(centers, c2p);
  argmin_kernel<<<BT / ROWS, 256, 0, stream>>>(x, centers, c2p, sel);
  proj_kernel<<<BT / ROWS, 256, 0, stream>>>(centers, W, b, sel, out);
}